// HIPABlock_42752104464975
// MI455X (gfx1250) — compile-verified
//
#include <hip/hip_runtime.h>
#include <math.h>

typedef float v2f __attribute__((ext_vector_type(2)));
typedef float v8f __attribute__((ext_vector_type(8)));

#define BATCH 4
#define CH    256
#define HH    512
#define WW    512
#define KTOT  13   // 1 (level0) + 4 (level1) + 8 (level2)

__device__ __forceinline__ float clipf(float x) {
    return fminf(fmaxf(x, 1e-8f), 20000.0f);
}

// ---------------------------------------------------------------------------
// Kernel 1: base of the max pyramid, g=4 -> pyr2[B][C][16]
// grid = B*C*4 blocks (one per (b,c,cell-row i)), 128 threads (4 cols each)
// Each wave (32 lanes) covers exactly one 128-wide cell -> shfl reduce.
// ---------------------------------------------------------------------------
__global__ __launch_bounds__(128)
void pool_kernel(const float* __restrict__ x, float* __restrict__ pyr2) {
    int blk = blockIdx.x;          // (b*C + c)*4 + i
    int i   = blk & 3;             // cell row (y block)
    int bc  = blk >> 2;            // b*C + c
    int t   = threadIdx.x;         // 0..127, handles columns 4t..4t+3
    const float4* p = (const float4*)(x + ((size_t)bc * HH + (size_t)i * 128) * WW);
    float4 m = p[t];
    #pragma unroll 4
    for (int r = 1; r < 128; ++r) {
        float4 v = p[(size_t)r * (WW / 4) + t];
        m.x = fmaxf(m.x, v.x); m.y = fmaxf(m.y, v.y);
        m.z = fmaxf(m.z, v.z); m.w = fmaxf(m.w, v.w);
    }
    float mm = fmaxf(fmaxf(m.x, m.y), fmaxf(m.z, m.w));
    #pragma unroll
    for (int off = 16; off > 0; off >>= 1)
        mm = fmaxf(mm, __shfl_xor(mm, off, 32));
    if ((t & 31) == 0)
        pyr2[(size_t)bc * 16 + i * 4 + (t >> 5)] = mm;   // cell n = i*4 + j
}

// ---------------------------------------------------------------------------
// Kernel 2: per-batch norms, importance, top-k, owner LUT, coords, LayerNorm.
// grid = B blocks, 256 threads (one per channel).
// ---------------------------------------------------------------------------
__global__ __launch_bounds__(256)
void score_kernel(const float* __restrict__ pyr2,
                  const float* __restrict__ gamma,
                  const float* __restrict__ beta,
                  float* __restrict__ ln_feats,    // [64][256] (rows b*13+s, pad zeroed)
                  int*   __restrict__ owner_cell,  // [B][16]
                  float* __restrict__ crd_out,     // [B][13][4]
                  float* __restrict__ sparsity_out) {
    const int b = blockIdx.x;
    const int c = threadIdx.x;
    __shared__ float red[256][17];
    __shared__ float s_n2sq[16], s_sum2[16];
    __shared__ float s_n1sq[4],  s_sum1[4];
    __shared__ float s_n0sq,     s_sum0;
    __shared__ int   s_lev[KTOT], s_idx[KTOT];

    float v[16];
    const float* rowp = pyr2 + ((size_t)b * CH + c) * 16;
    #pragma unroll
    for (int n = 0; n < 16; ++n) v[n] = rowp[n];

    // level-1 (2x2) and level-0 maxima for this channel
    float p1[4];
    #pragma unroll
    for (int qi = 0; qi < 2; ++qi)
        #pragma unroll
        for (int qj = 0; qj < 2; ++qj) {
            int n0 = (qi * 2) * 4 + qj * 2;   // top-left cell of quadrant
            p1[qi * 2 + qj] = fmaxf(fmaxf(v[n0], v[n0 + 1]),
                                    fmaxf(v[n0 + 4], v[n0 + 5]));
        }
    float p0 = fmaxf(fmaxf(p1[0], p1[1]), fmaxf(p1[2], p1[3]));

    auto reduce = [&](int nv) {
        __syncthreads();
        for (int s = 128; s >= 1; s >>= 1) {
            if (c < s)
                for (int n = 0; n < nv; ++n) red[c][n] += red[c + s][n];
            __syncthreads();
        }
    };

    // pass A: sum of squares per level-2 cell
    #pragma unroll
    for (int n = 0; n < 16; ++n) red[c][n] = v[n] * v[n];
    reduce(16);
    if (c < 16) s_n2sq[c] = red[0][c];
    __syncthreads();

    // pass B: plain sums per level-2 cell (for LayerNorm mean)
    #pragma unroll
    for (int n = 0; n < 16; ++n) red[c][n] = v[n];
    reduce(16);
    if (c < 16) s_sum2[c] = red[0][c];
    __syncthreads();

    // pass C: level-1 / level-0 sumsq + sums
    #pragma unroll
    for (int q = 0; q < 4; ++q) { red[c][q] = p1[q] * p1[q]; red[c][4 + q] = p1[q]; }
    red[c][8] = p0 * p0; red[c][9] = p0;
    reduce(10);
    if (c < 4) { s_n1sq[c] = red[0][c]; s_sum1[c] = red[0][4 + c]; }
    if (c == 0) { s_n0sq = red[0][8]; s_sum0 = red[0][9]; }
    __syncthreads();

    if (c == 0) {
        float n2[16], n1[4];
        for (int n = 0; n < 16; ++n) n2[n] = sqrtf(s_n2sq[n]);
        for (int q = 0; q < 4;  ++q) n1[q] = sqrtf(s_n1sq[q]);
        float n0 = sqrtf(s_n0sq);

        float imp2[16], imp1[4];
        for (int n = 0; n < 16; ++n) imp2[n] = clipf(n2[n]);  // top level: norm itself
        for (int q = 0; q < 4; ++q) {
            int qi = q >> 1, qj = q & 1;
            int par = (2 * qi) * 4 + (2 * qj);                 // parent[::2,::2]
            imp1[q] = clipf(fabsf(n1[q] - n2[par]));
        }
        // (level 0: N=1, k=1 -> trivially index 0)

        // top-k (selection, strict '>' => lax.top_k tie-break: lowest index first)
        int kept2[8]; bool u2[16];
        for (int n = 0; n < 16; ++n) u2[n] = false;
        for (int r = 0; r < 8; ++r) {
            int best = -1; float bv = -1e30f;
            for (int n = 0; n < 16; ++n)
                if (!u2[n] && imp2[n] > bv) { bv = imp2[n]; best = n; }
            u2[best] = true; kept2[r] = best;
        }
        int kept1[4]; bool u1[4] = {false, false, false, false};
        for (int r = 0; r < 4; ++r) {
            int best = -1; float bv = -1e30f;
            for (int q = 0; q < 4; ++q)
                if (!u1[q] && imp1[q] > bv) { bv = imp1[q]; best = q; }
            u1[best] = true; kept1[r] = best;
        }

        s_lev[0] = 0; s_idx[0] = 0;
        for (int r = 0; r < 4; ++r) { s_lev[1 + r] = 1; s_idx[1 + r] = kept1[r]; }
        for (int r = 0; r < 8; ++r) { s_lev[5 + r] = 2; s_idx[5 + r] = kept2[r]; }

        // owner LUT: level-2 box (idx 5..12) wins if kept, else quadrant box (1..4)
        int qrank[4];
        for (int r = 0; r < 4; ++r) qrank[kept1[r]] = 1 + r;
        int own[16];
        for (int n = 0; n < 16; ++n) {
            int qi = (n >> 2) >> 1, qj = (n & 3) >> 1;
            own[n] = qrank[qi * 2 + qj];
        }
        for (int r = 0; r < 8; ++r) own[kept2[r]] = 5 + r;
        for (int n = 0; n < 16; ++n) owner_cell[b * 16 + n] = own[n];

        // coords output
        for (int s = 0; s < KTOT; ++s) {
            int g  = 1 << s_lev[s];
            int gi = s_idx[s] / g, gj = s_idx[s] % g;
            float* cp = crd_out + ((size_t)b * KTOT + s) * 4;
            cp[0] = (gj + 0.5f) / g;  // cx
            cp[1] = (gi + 0.5f) / g;  // cy
            cp[2] = 1.0f / g;         // w
            cp[3] = 1.0f / g;         // h
        }
        if (b == 0) sparsity_out[0] = (float)KTOT / (float)(HH * WW);
    }
    __syncthreads();

    // LayerNorm, one row per kept entry
    float gc = gamma[c], bc_ = beta[c];
    for (int s = 0; s < KTOT; ++s) {
        int lev = s_lev[s], idx = s_idx[s];
        float f, sum, sq;
        if (lev == 2)      { f = v[idx];  sum = s_sum2[idx]; sq = s_n2sq[idx]; }
        else if (lev == 1) { f = p1[idx]; sum = s_sum1[idx]; sq = s_n1sq[idx]; }
        else               { f = p0;      sum = s_sum0;      sq = s_n0sq;      }
        float mu  = sum * (1.0f / CH);
        float var = sq  * (1.0f / CH) - mu * mu;
        ln_feats[((size_t)b * KTOT + s) * CH + c] =
            (f - mu) * rsqrtf(var + 1e-5f) * gc + bc_;
    }
    if (b == 0)
        for (int r = BATCH * KTOT; r < 64; ++r)
            ln_feats[(size_t)r * CH + c] = 0.0f;   // pad rows for WMMA tiles
}

// ---------------------------------------------------------------------------
// Kernel 3: seq = LN(feats) @ w_proj + b_proj, full-f32 WMMA (16x16x4).
// grid = 64 blocks (4 row-tiles x 16 col-tiles), 32 threads (one wave).
// A 16x4 layout: lane l -> A[l&15][2*(l>>4)+v]; B 4x16: lane l -> B[2*(l>>4)+v][l&15]
// D 16x16: lane l VGPR v -> D[8*(l>>4)+v][l&15]   (ISA 7.12.2)
// ---------------------------------------------------------------------------
__global__ __launch_bounds__(32)
void gemm_wmma_kernel(const float* __restrict__ A,    // [64][256] LN feats (padded)
                      const float* __restrict__ Wp,   // [256][256] row-major (K x N)
                      const float* __restrict__ bias, // [256]
                      float* __restrict__ seq_out) {  // [52][256]
    int rt = blockIdx.x >> 4;        // row tile 0..3
    int ct = blockIdx.x & 15;        // col tile 0..15
    int l    = threadIdx.x;          // 0..31
    int half = l >> 4;               // 0/1
    int lm   = l & 15;
    int rowA = rt * 16 + lm;         // A-matrix row this lane feeds
    int coln = ct * 16 + lm;         // B/D column this lane holds

    v8f acc = {};
    for (int k0 = 0; k0 < CH; k0 += 4) {
        int ka = k0 + half * 2;
        v2f a = *(const v2f*)(A + (size_t)rowA * CH + ka);          // A[m][ka..ka+1]
        v2f bb;
        bb.x = Wp[(size_t)ka * CH + coln];                          // B[ka][n]
        bb.y = Wp[(size_t)(ka + 1) * CH + coln];                    // B[ka+1][n]
        acc = __builtin_amdgcn_wmma_f32_16x16x4_f32(
            /*neg_a=*/false, a, /*neg_b=*/false, bb,
            /*c_mod=*/(short)0, acc, /*reuse_a=*/false, /*reuse_b=*/false);
    }
    float bv = bias[coln];
    #pragma unroll
    for (int vv = 0; vv < 8; ++vv) {
        int m = rt * 16 + half * 8 + vv;           // global output row
        if (m < BATCH * KTOT)
            seq_out[(size_t)m * CH + coln] = acc[vv] + bv;
    }
}

// ---------------------------------------------------------------------------
// Kernel 4: box fill. out_sparse[b,c,y,x] = seq[b, owner(cell(y,x)), c].
// Level-0/1 boxes cover everything, so owner >= 0 always. float4 stores.
// 2^26 float4s: grid 262144 x 256.
// ---------------------------------------------------------------------------
__global__ __launch_bounds__(256)
void fill_kernel(const float* __restrict__ seq,        // [B][13][256]
                 const int*   __restrict__ owner_cell, // [B][16]
                 float* __restrict__ out) {            // [B][256][512][512]
    size_t idx4 = (size_t)blockIdx.x * blockDim.x + threadIdx.x;
    int x4 = (int)(idx4 & 127);          // x/4
    int y  = (int)((idx4 >> 7) & 511);
    int c  = (int)((idx4 >> 16) & 255);
    int b  = (int)(idx4 >> 24);
    int cell = ((y >> 7) << 2) | (x4 >> 5);
    int own  = owner_cell[b * 16 + cell];
    float val = seq[((size_t)b * KTOT + own) * CH + c];
    float4 o = {val, val, val, val};
    ((float4*)out)[idx4] = o;
}

// ---------------------------------------------------------------------------
extern "C" void kernel_launch(void* const* d_in, const int* in_sizes, int n_in,
                              void* d_out, int out_size, void* d_ws, size_t ws_size,
                              hipStream_t stream) {
    const float* x     = (const float*)d_in[0];
    const float* gamma = (const float*)d_in[1];
    const float* beta  = (const float*)d_in[2];
    const float* wp    = (const float*)d_in[3];
    const float* bp    = (const float*)d_in[4];

    float* out        = (float*)d_out;
    float* out_sparse = out;                                     // B*Cout*H*W
    float* out_seq    = out + (size_t)BATCH * CH * HH * WW;      // B*13*256
    float* out_crd    = out_seq + (size_t)BATCH * KTOT * CH;     // B*13*4
    float* out_sp     = out_crd + (size_t)BATCH * KTOT * 4;      // 1

    float* ws      = (float*)d_ws;
    float* ws_pyr2 = ws;                 // B*C*16 = 16384 floats
    float* ws_ln   = ws + 16384;         // 64*256 = 16384 floats
    int*   ws_own  = (int*)(ws + 32768); // B*16 ints

    pool_kernel<<<BATCH * CH * 4, 128, 0, stream>>>(x, ws_pyr2);
    score_kernel<<<BATCH, 256, 0, stream>>>(ws_pyr2, gamma, beta,
                                            ws_ln, ws_own, out_crd, out_sp);
    gemm_wmma_kernel<<<64, 32, 0, stream>>>(ws_ln, wp, bp, out_seq);
    fill_kernel<<<(BATCH * CH * HH * WW / 4) / 256, 256, 0, stream>>>(
        out_seq, ws_own, out_sparse);
}